// QuantumLayer_61907658604924
// MI455X (gfx1250) — compile-verified
//
#include <hip/hip_runtime.h>

// ---------------------------------------------------------------------------
// Quantum layer (4 qubits, DIM=16, 3 layers) for MI455X / gfx1250.
//
// Pipeline:
//   prep kernel (1 block): folds the 12 RX gates + 12 CNOT permutations into
//     one dense 16x16 complex matrix W = M2*M1*M0, written to d_ws.
//   main kernel: RY on |0> is a product state (24 muls). The batched state
//     transform S(16x16) -> S*W^T (S real => two real matmuls) and the final
//     probs @ Z_SIGNS contraction run on V_WMMA_F32_16X16X4_F32, K=16 as
//     4 chunks of K=4. 24 WMMAs per wave of 32 batch elements.
// ---------------------------------------------------------------------------

typedef float v2f __attribute__((ext_vector_type(2)));
typedef float v8f __attribute__((ext_vector_type(8)));

#define SROW 20   // padded LDS row stride (floats): 80B rows -> 16B-aligned,
                  // distinct start banks for 16 consecutive rows

__device__ __forceinline__ int cnot_perm_apply(int q, int i) {
    // reference: control c=q, target t=(q+1)%4; qubit q <-> bit (3-q)
    int t = (q + 1) & 3;
    int cb = (i >> (3 - q)) & 1;
    return cb ? (i ^ (1 << (3 - t))) : i;
}

__device__ __forceinline__ int layer_perm(int i) {
    // state after the 4 CNOT gathers: s'[i] = s[p0(p1(p2(p3(i))))]
    return cnot_perm_apply(0, cnot_perm_apply(1, cnot_perm_apply(2, cnot_perm_apply(3, i))));
}

// ---------------------------------------------------------------------------
// Prep: build W (16x16 complex) from the 12 weights. One block of 256.
// Layout in d_ws: Wg[0..255] = Re(W) row-major, Wg[256..511] = Im(W).
// ---------------------------------------------------------------------------
__global__ __launch_bounds__(256)
void qc_prep_kernel(const float* __restrict__ weights, float* __restrict__ Wg)
{
    __shared__ float sCSc[12], sCSs[12];
    __shared__ float sMr[3 * 256], sMi[3 * 256];
    __shared__ float sTr[256], sTi[256];

    const int t = threadIdx.x;
    if (t < 12) {
        float w  = weights[t];
        sCSc[t] = cosf(0.5f * w);
        sCSs[t] = sinf(0.5f * w);
    }
    __syncthreads();

    const int i = t >> 4, j = t & 15;
    const int r = layer_perm(i);
    // M_l[i][j] = (RX tensor)[P(i)][j]
    #pragma unroll
    for (int l = 0; l < 3; ++l) {
        float ur = 1.0f, ui = 0.0f;            // product of 4 complex factors
        #pragma unroll
        for (int q = 0; q < 4; ++q) {
            const float c = sCSc[l * 4 + q];
            const float s = sCSs[l * 4 + q];
            if (((r >> (3 - q)) & 1) == ((j >> (3 - q)) & 1)) {
                ur *= c; ui *= c;              // diagonal RX element (c, 0)
            } else {                           // off-diagonal (0, -s)
                const float tmp = ur;
                ur = s * ui;
                ui = -s * tmp;
            }
        }
        sMr[l * 256 + i * 16 + j] = ur;
        sMi[l * 256 + i * 16 + j] = ui;
    }
    __syncthreads();

    // T = M1 * M0
    float tr = 0.0f, ti = 0.0f;
    #pragma unroll
    for (int k = 0; k < 16; ++k) {
        const float ar = sMr[256 + i * 16 + k], ai = sMi[256 + i * 16 + k];
        const float br = sMr[k * 16 + j],       bi = sMi[k * 16 + j];
        tr += ar * br - ai * bi;
        ti += ar * bi + ai * br;
    }
    sTr[i * 16 + j] = tr;
    sTi[i * 16 + j] = ti;
    __syncthreads();

    // W = M2 * T
    float wr = 0.0f, wi = 0.0f;
    #pragma unroll
    for (int k = 0; k < 16; ++k) {
        const float ar = sMr[512 + i * 16 + k], ai = sMi[512 + i * 16 + k];
        const float br = sTr[k * 16 + j],       bi = sTi[k * 16 + j];
        wr += ar * br - ai * bi;
        wi += ar * bi + ai * br;
    }
    Wg[t]       = wr;
    Wg[256 + t] = wi;
}

// ---------------------------------------------------------------------------
// Main WMMA kernel: 256 threads = 8 waves, one batch element per lane.
// ---------------------------------------------------------------------------
__global__ __launch_bounds__(256)
void qc_wmma_kernel(const float4* __restrict__ inputs,
                    const float*  __restrict__ Wg,
                    float4*       __restrict__ out)
{
    __shared__ float sWr[16 * SROW], sWi[16 * SROW]; // combined layer matrix
    __shared__ float sS[8 * 32 * SROW];              // per-wave post-RY states
    __shared__ float sP[8 * 16 * SROW];              // per-wave probs staging

    const int t    = threadIdx.x;
    const int lane = t & 31;
    const int wave = t >> 5;
    const int gid  = blockIdx.x * 256 + t;

    // cooperative load of W into padded LDS (2 floats / thread)
    {
        const int i = t >> 4, j = t & 15;
        sWr[i * SROW + j] = Wg[t];
        sWi[i * SROW + j] = Wg[256 + t];
    }

    // ---- RY on |0> is a product state: re[i] = prod_q f_q(bit_q(i)) ------
    const float4 th = inputs[gid];
    const float c0 = __cosf(0.5f * th.x), s0 = __sinf(0.5f * th.x);
    const float c1 = __cosf(0.5f * th.y), s1 = __sinf(0.5f * th.y);
    const float c2 = __cosf(0.5f * th.z), s2 = __sinf(0.5f * th.z);
    const float c3 = __cosf(0.5f * th.w), s3 = __sinf(0.5f * th.w);
    // i bits (MSB..LSB) = qubits 0..3 ; re[i] = a01[i>>2] * a23[i&3]
    const float a01[4] = { c0 * c1, c0 * s1, s0 * c1, s0 * s1 };
    const float a23[4] = { c2 * c3, c2 * s3, s2 * c3, s2 * s3 };

    float* myrow = &sS[(wave * 32 + lane) * SROW];
    #pragma unroll
    for (int hi = 0; hi < 4; ++hi) {
        *(float4*)&myrow[4 * hi] = make_float4(a01[hi] * a23[0], a01[hi] * a23[1],
                                               a01[hi] * a23[2], a01[hi] * a23[3]);
    }

    __syncthreads();   // publishes sWr/sWi across waves (sS is intra-wave)

    // WMMA f32 16x16x4 operand lane roles (ISA 7.12.2):
    //   A: M = lane&15; v0/v1 = K{0,1} (lanes<16) or K{2,3} (lanes>=16)
    //   B: N = lane&15; v0/v1 = K{0,1} (lanes<16) or K{2,3} (lanes>=16)
    const int n  = lane & 15;
    const int kb = (lane < 16) ? 0 : 2;

    // B operands for the state transform: B_k[kl][n] = W[n][4k+kl]
    v2f Br[4], Bi[4];
    #pragma unroll
    for (int k = 0; k < 4; ++k) {
        Br[k] = *(const v2f*)&sWr[n * SROW + 4 * k + kb];
        Bi[k] = *(const v2f*)&sWi[n * SROW + 4 * k + kb];
    }

    v8f D2g0, D2g1;
    #pragma unroll
    for (int g = 0; g < 2; ++g) {              // two 16-element batch groups
        // --- state transform: Re = S*B(Wr), Im = S*B(Wi)  (S real) --------
        v8f Dre = {}; v8f Dim = {};
        #pragma unroll
        for (int k = 0; k < 4; ++k) {
            const v2f a = *(const v2f*)&sS[(wave * 32 + 16 * g + n) * SROW + 4 * k + kb];
            Dre = __builtin_amdgcn_wmma_f32_16x16x4_f32(false, a, false, Br[k],
                                                        (short)0, Dre, false, false);
            Dim = __builtin_amdgcn_wmma_f32_16x16x4_f32(false, a, false, Bi[k],
                                                        (short)0, Dim, false, false);
        }
        // --- probs in C/D layout: P[m][n'] = re^2 + im^2 -------------------
        v8f P;
        #pragma unroll
        for (int j = 0; j < 8; ++j) P[j] = Dre[j] * Dre[j] + Dim[j] * Dim[j];

        // intra-wave staging: DS ops of one wave execute in order, so no
        // barrier needed; fence stops IR-level reordering.
        __asm__ volatile("" ::: "memory");
        const int rbase = (lane < 16) ? 0 : 8; // C/D layout row mapping
        #pragma unroll
        for (int j = 0; j < 8; ++j)
            sP[(wave * 16 + rbase + j) * SROW + n] = P[j];
        __asm__ volatile("" ::: "memory");

        // --- out^T = Z^T * P^T with constant +-1 A operand -----------------
        // Z^T occupies A rows 0-3 (g=0) or 8-11 (g=1) so group-0 results
        // land in lanes 0-15 and group-1 results in lanes 16-31.
        v8f D2 = {};
        #pragma unroll
        for (int k = 0; k < 4; ++k) {
            const v2f b2 = *(const v2f*)&sP[(wave * 16 + n) * SROW + 4 * k + kb];
            const bool act = ((n >> 2) == (g << 1));
            const int  qz  = n & 3;
            const int  i0  = 4 * k + kb;
            v2f a2;
            a2.x = act ? (1.0f - 2.0f * (float)((i0       >> (3 - qz)) & 1)) : 0.0f;
            a2.y = act ? (1.0f - 2.0f * (float)(((i0 + 1) >> (3 - qz)) & 1)) : 0.0f;
            D2 = __builtin_amdgcn_wmma_f32_16x16x4_f32(false, a2, false, b2,
                                                       (short)0, D2, false, false);
        }
        if (g == 0) D2g0 = D2; else D2g1 = D2;
        __asm__ volatile("" ::: "memory");     // order sP reuse (WAR) next g
    }

    // lane<16: element from D2g0[0..3]; lane>=16: from D2g1[0..3]
    const bool lo = (lane < 16);
    float4 o;
    o.x = lo ? D2g0[0] : D2g1[0];
    o.y = lo ? D2g0[1] : D2g1[1];
    o.z = lo ? D2g0[2] : D2g1[2];
    o.w = lo ? D2g0[3] : D2g1[3];
    out[gid] = o;
}

// ---------------------------------------------------------------------------
// Scalar fallback / tail kernel (tail only if B % 256 != 0; full fallback
// only if ws_size < 2KB — neither happens with the reference shapes).
// ---------------------------------------------------------------------------
__global__ void qc_scalar_kernel(const float4* __restrict__ inputs,
                                 const float*  __restrict__ weights,
                                 float4*       __restrict__ out,
                                 int base, int B)
{
    const int e = base + blockIdx.x * blockDim.x + threadIdx.x;
    if (e >= B) return;

    float sr[16], si[16];
    #pragma unroll
    for (int i = 0; i < 16; ++i) { sr[i] = 0.0f; si[i] = 0.0f; }

    const float4 th = inputs[e];
    {
        const float c0 = __cosf(0.5f * th.x), s0 = __sinf(0.5f * th.x);
        const float c1 = __cosf(0.5f * th.y), s1 = __sinf(0.5f * th.y);
        const float c2 = __cosf(0.5f * th.z), s2 = __sinf(0.5f * th.z);
        const float c3 = __cosf(0.5f * th.w), s3 = __sinf(0.5f * th.w);
        const float a01[4] = { c0 * c1, c0 * s1, s0 * c1, s0 * s1 };
        const float a23[4] = { c2 * c3, c2 * s3, s2 * c3, s2 * s3 };
        #pragma unroll
        for (int i = 0; i < 16; ++i) sr[i] = a01[i >> 2] * a23[i & 3];
    }

    for (int l = 0; l < 3; ++l) {
        #pragma unroll
        for (int q = 0; q < 4; ++q) {
            const float w = weights[l * 4 + q];
            const float c = __cosf(0.5f * w);
            const float s = __sinf(0.5f * w);
            const int   m = 1 << (3 - q);
            #pragma unroll
            for (int i0 = 0; i0 < 16; ++i0) {
                if (i0 & m) continue;
                const int i1 = i0 | m;
                const float r0 = sr[i0], i0v = si[i0];
                const float r1 = sr[i1], i1v = si[i1];
                sr[i0] = c * r0 + s * i1v;
                si[i0] = c * i0v - s * r1;
                sr[i1] = s * i0v + c * r1;
                si[i1] = -s * r0 + c * i1v;
            }
        }
        float nr[16], ni[16];
        #pragma unroll
        for (int i = 0; i < 16; ++i) {
            const int p = layer_perm(i);
            nr[i] = sr[p]; ni[i] = si[p];
        }
        #pragma unroll
        for (int i = 0; i < 16; ++i) { sr[i] = nr[i]; si[i] = ni[i]; }
    }

    float o[4] = {0.0f, 0.0f, 0.0f, 0.0f};
    #pragma unroll
    for (int i = 0; i < 16; ++i) {
        const float p = sr[i] * sr[i] + si[i] * si[i];
        #pragma unroll
        for (int q = 0; q < 4; ++q) {
            o[q] += p * (1.0f - 2.0f * (float)((i >> (3 - q)) & 1));
        }
    }
    out[e] = make_float4(o[0], o[1], o[2], o[3]);
}

extern "C" void kernel_launch(void* const* d_in, const int* in_sizes, int n_in,
                              void* d_out, int out_size, void* d_ws, size_t ws_size,
                              hipStream_t stream)
{
    (void)n_in; (void)out_size;
    const float4* inputs  = (const float4*)d_in[0];   // [B][4] float32
    const float*  weights = (const float*)d_in[1];    // [3][4] float32
    float4*       out     = (float4*)d_out;           // [B][4] float32
    const int B = in_sizes[0] / 4;

    if (ws_size >= 512 * sizeof(float)) {
        float* Wg = (float*)d_ws;                     // 16x16 complex W
        qc_prep_kernel<<<1, 256, 0, stream>>>(weights, Wg);

        const int fullBlocks = B >> 8;                // 256 elements per block
        if (fullBlocks > 0) {
            qc_wmma_kernel<<<fullBlocks, 256, 0, stream>>>(inputs, Wg, out);
        }
        const int rem = B & 255;
        if (rem > 0) {
            qc_scalar_kernel<<<(rem + 255) / 256, 256, 0, stream>>>(
                inputs, weights, out, fullBlocks * 256, B);
        }
    } else {
        // scratch too small (never expected): fully scalar but correct
        qc_scalar_kernel<<<(B + 255) / 256, 256, 0, stream>>>(
            inputs, weights, out, 0, B);
    }
}